// GCNMultiLabelClassifier_7490422964950
// MI455X (gfx1250) — compile-verified
//
#include <hip/hip_runtime.h>
#include <hip/hip_bf16.h>

#define N_NODES 100000
#define N_EDGES 1600000
#define IN_DIM  256
#define HID     128
#define N_GRAPHS 64
#define OUT_MF 489
#define OUT_BP 1943
#define OUT_CC 320

typedef __attribute__((ext_vector_type(2))) float v2f;
typedef __attribute__((ext_vector_type(8))) float v8f;

// ---------------------------------------------------------------- utilities
__global__ void fill_f32(float* __restrict__ p, float v, int n) {
    int t = blockIdx.x * blockDim.x + threadIdx.x;
    if (t < n) p[t] = v;
}

__global__ void zero_f4(float4* __restrict__ p, int n4) {
    int t = blockIdx.x * blockDim.x + threadIdx.x;
    if (t < n4) { float4 z; z.x = z.y = z.z = z.w = 0.f; p[t] = z; }
}

__global__ void deg_accum(const int* __restrict__ dst, float* __restrict__ deg, int E) {
    int t = blockIdx.x * blockDim.x + threadIdx.x;
    if (t < E) atomicAdd(&deg[dst[t]], 1.0f);
}

__global__ void rsqrt_inplace(float* __restrict__ p, int n) {
    int t = blockIdx.x * blockDim.x + threadIdx.x;
    if (t < n) p[t] = rsqrtf(p[t]);
}

// ---------------------------------------------------------------- WMMA GEMM
// out[M x 128] = A[M x K] @ W[K x 128], fp32 exact via V_WMMA_F32_16X16X4_F32.
// grid.x = M/16 row tiles; 256 threads = 8 waves, wave w owns col tile w.
// W is staged through LDS in 64-row K-chunks (32 KB).
template <int K>
__global__ __launch_bounds__(256) void gemm_wmma_f32(const float* __restrict__ A,
                                                     const float* __restrict__ W,
                                                     float* __restrict__ out) {
    __shared__ float ldsW[64 * HID];   // 32 KB chunk of W
    const int tid  = threadIdx.x;
    const int wave = tid >> 5;         // 0..7 -> column tile
    const int lane = tid & 31;
    const int half = lane >> 4;        // K sub-select (0: K=0,1  1: K=2,3)
    const int m    = lane & 15;        // row (A) / col (B,D) within tile
    const int row0 = blockIdx.x * 16;
    const int col0 = wave * 16;

    const float* arow = A + (size_t)(row0 + m) * K;
    v8f acc = {};

    for (int kb = 0; kb < K; kb += 64) {
        // cooperative load of W rows [kb, kb+64) into LDS
        const float4* wv = (const float4*)(W + (size_t)kb * HID);
        float4* lv = (float4*)ldsW;
        #pragma unroll
        for (int i = tid; i < 64 * HID / 4; i += 256) lv[i] = wv[i];
        __syncthreads();

        #pragma unroll
        for (int k = 0; k < 64; k += 4) {
            const int ka = k + 2 * half;
            v2f a;                                   // A 16x4: lane half picks K pair
            a.x = arow[kb + ka];
            a.y = arow[kb + ka + 1];
            v2f b;                                   // B 4x16: row ka, col col0+m
            b.x = ldsW[ka * HID + col0 + m];
            b.y = ldsW[(ka + 1) * HID + col0 + m];
            acc = __builtin_amdgcn_wmma_f32_16x16x4_f32(
                false, a, false, b, (short)0, acc, false, false);
        }
        __syncthreads();
    }

    // D 16x16 fp32: VGPR v -> row v + 8*half, col col0 + m
    #pragma unroll
    for (int v = 0; v < 8; ++v)
        out[(size_t)(row0 + v + 8 * half) * HID + col0 + m] = acc[v];
}

// ---------------------------------------------------------------- scatter
// agg[dst] += h[src] * dis[src]*dis[dst] ; one wave per edge, lane = feature.
__global__ __launch_bounds__(256) void edge_scatter(const int* __restrict__ src,
                                                    const int* __restrict__ dst,
                                                    const float* __restrict__ dis,
                                                    const float* __restrict__ h,
                                                    float* __restrict__ agg, int E) {
    const int lane = threadIdx.x & 31;
    const int gw   = (blockIdx.x * blockDim.x + threadIdx.x) >> 5;
    const int nw   = (gridDim.x * blockDim.x) >> 5;
    for (int e = gw; e < E; e += nw) {
        const int s = src[e], d = dst[e];
        const float nrm = dis[s] * dis[d];
        const float* hs = h + (size_t)s * HID;
        float* ad = agg + (size_t)d * HID;
        #pragma unroll
        for (int j = 0; j < HID; j += 32)
            atomicAdd(&ad[j + lane], hs[j + lane] * nrm);
    }
}

// agg = relu(agg + dis^2 * h + bias), in place
__global__ void finish_layer(float* __restrict__ agg, const float* __restrict__ h,
                             const float* __restrict__ dis, const float* __restrict__ bias,
                             int n) {
    int t = blockIdx.x * blockDim.x + threadIdx.x;
    if (t >= n) return;
    const int i = t >> 7, j = t & (HID - 1);
    const float di = dis[i];
    const float v = agg[t] + di * di * h[t] + bias[j];
    agg[t] = v > 0.f ? v : 0.f;
}

// ---------------------------------------------------------------- pooling
__global__ __launch_bounds__(256) void pool_accum(const float* __restrict__ h,
                                                  const int* __restrict__ batch,
                                                  float* __restrict__ pooled,
                                                  float* __restrict__ cnt, int N) {
    const int lane = threadIdx.x & 31;
    const int gw   = (blockIdx.x * blockDim.x + threadIdx.x) >> 5;
    const int nw   = (gridDim.x * blockDim.x) >> 5;
    for (int i = gw; i < N; i += nw) {
        const int g = batch[i];
        const float* hi = h + (size_t)i * HID;
        float* pg = pooled + g * HID;
        #pragma unroll
        for (int j = 0; j < HID; j += 32)
            atomicAdd(&pg[j + lane], hi[j + lane]);
        if (lane == 0) atomicAdd(&cnt[g], 1.0f);
    }
}

__global__ void pool_div(float* __restrict__ pooled, const float* __restrict__ cnt) {
    int t = blockIdx.x * blockDim.x + threadIdx.x;
    if (t >= N_GRAPHS * HID) return;
    float c = cnt[t >> 7];
    pooled[t] /= (c > 1.f ? c : 1.f);
}

// ---------------------------------------------------------------- heads
__global__ void head_gemm(const float* __restrict__ pooled, const float* __restrict__ W,
                          const float* __restrict__ b, float* __restrict__ out, int OUT) {
    int t = blockIdx.x * blockDim.x + threadIdx.x;
    const int n = N_GRAPHS * OUT;
    if (t >= n) return;
    const int g = t / OUT, o = t - g * OUT;
    const float* p = pooled + g * HID;
    float s = b[o];
    #pragma unroll 8
    for (int k = 0; k < HID; ++k) s += p[k] * W[k * OUT + o];
    out[t] = s;
}

// ---------------------------------------------------------------- launch
extern "C" void kernel_launch(void* const* d_in, const int* in_sizes, int n_in,
                              void* d_out, int out_size, void* d_ws, size_t ws_size,
                              hipStream_t stream) {
    const float* x   = (const float*)d_in[0];
    const int*   ei  = (const int*)d_in[1];
    const int*   bat = (const int*)d_in[2];
    const float* W1  = (const float*)d_in[3];
    const float* b1  = (const float*)d_in[4];
    const float* W2  = (const float*)d_in[5];
    const float* b2  = (const float*)d_in[6];
    const float* Wmf = (const float*)d_in[7];
    const float* bmf = (const float*)d_in[8];
    const float* Wbp = (const float*)d_in[9];
    const float* bbp = (const float*)d_in[10];
    const float* Wcc = (const float*)d_in[11];
    const float* bcc = (const float*)d_in[12];
    float* out = (float*)d_out;

    const int E = in_sizes[1] / 2;            // 1.6M
    const int* src = ei;
    const int* dst = ei + E;

    // workspace layout (floats); all offsets 16B aligned
    float* ws     = (float*)d_ws;
    float* dis    = ws;                                   // N
    float* h      = dis + N_NODES;                        // N*128
    float* agg    = h + (size_t)N_NODES * HID;            // N*128
    float* pooled = agg + (size_t)N_NODES * HID;          // 64*128
    float* cnt    = pooled + N_GRAPHS * HID;              // 64

    const int NH   = N_NODES * HID;                       // 12.8M
    const int NH4  = NH / 4;

    // 1) degree (with self loop) -> rsqrt
    fill_f32<<<(N_NODES + 255) / 256, 256, 0, stream>>>(dis, 1.0f, N_NODES);
    deg_accum<<<(E + 255) / 256, 256, 0, stream>>>(dst, dis, E);
    rsqrt_inplace<<<(N_NODES + 255) / 256, 256, 0, stream>>>(dis, N_NODES);

    // 2) layer 1: h = x @ W1 (WMMA), scatter into agg, finish in agg
    gemm_wmma_f32<IN_DIM><<<N_NODES / 16, 256, 0, stream>>>(x, W1, h);
    zero_f4<<<(NH4 + 255) / 256, 256, 0, stream>>>((float4*)agg, NH4);
    edge_scatter<<<2048, 256, 0, stream>>>(src, dst, dis, h, agg, E);
    finish_layer<<<(NH + 255) / 256, 256, 0, stream>>>(agg, h, dis, b1, NH);
    // agg now holds h1act

    // 3) layer 2: h = h1act @ W2 (WMMA), scatter into (reset) agg-slot = h buf
    gemm_wmma_f32<HID><<<N_NODES / 16, 256, 0, stream>>>(agg, W2, h);
    // swap roles: pre-activation now in h; reuse agg buffer? it holds h1act which
    // is no longer needed -> zero it and scatter h into it.
    zero_f4<<<(NH4 + 255) / 256, 256, 0, stream>>>((float4*)agg, NH4);
    // need a scratch for agg2: we just zeroed agg (h1act dead after gemm2)
    edge_scatter<<<2048, 256, 0, stream>>>(src, dst, dis, h, agg, E);
    finish_layer<<<(NH + 255) / 256, 256, 0, stream>>>(agg, h, dis, b2, NH);
    // agg now holds h2act

    // 4) global mean pool
    fill_f32<<<(N_GRAPHS * HID + N_GRAPHS + 255) / 256, 256, 0, stream>>>(
        pooled, 0.0f, N_GRAPHS * HID + N_GRAPHS);
    pool_accum<<<12500, 256, 0, stream>>>(agg, bat, pooled, cnt, N_NODES);
    pool_div<<<(N_GRAPHS * HID + 255) / 256, 256, 0, stream>>>(pooled, cnt);

    // 5) heads (concatenated into d_out: mf | bp | cc)
    float* out_mf = out;
    float* out_bp = out + N_GRAPHS * OUT_MF;
    float* out_cc = out + N_GRAPHS * (OUT_MF + OUT_BP);
    head_gemm<<<(N_GRAPHS * OUT_MF + 255) / 256, 256, 0, stream>>>(pooled, Wmf, bmf, out_mf, OUT_MF);
    head_gemm<<<(N_GRAPHS * OUT_BP + 255) / 256, 256, 0, stream>>>(pooled, Wbp, bbp, out_bp, OUT_BP);
    head_gemm<<<(N_GRAPHS * OUT_CC + 255) / 256, 256, 0, stream>>>(pooled, Wcc, bcc, out_cc, OUT_CC);
}